// DecorrLinear_20340965113927
// MI455X (gfx1250) — compile-verified
//
#include <hip/hip_runtime.h>
#include <hip/hip_bf16.h>

typedef __attribute__((ext_vector_type(2))) float v2f;
typedef __attribute__((ext_vector_type(4))) float v4f;
typedef __attribute__((ext_vector_type(8))) float v8f;

constexpr int D  = 1024;     // model dim
constexpr int L  = 8192;     // sequence length
constexpr int NBATCH = 4;
constexpr int NS = 2048;     // samples per batch
constexpr int MY = NBATCH * L;   // 32768 rows of y
constexpr int NX = NBATCH * NS;  // 8192 sampled rows
constexpr float KAPPA = 0.5f;

__device__ __forceinline__ v8f wmma_f32(v2f a, v2f b, v8f c) {
  // D = A(16x4, f32) x B(4x16, f32) + C(16x16, f32)
  return __builtin_amdgcn_wmma_f32_16x16x4_f32(
      /*neg_a=*/false, a, /*neg_b=*/false, b,
      /*c_mod=*/(short)0, c, /*reuse_a=*/false, /*reuse_b=*/false);
}

// ---------------------------------------------------------------------------
// C[M x D] = A[M x D] * B[D x D]^T (+bias), optional row gather on A.
// One wave computes a 64x32 tile: MREP=4 x NREP=2, so each A fragment feeds
// 2 WMMAs and each B fragment feeds 4.  Per k-step: 4 A + 2 B v2f loads for
// 8 WMMAs (0.75 loads/WMMA).
// A-fragment lane l: A[m16 + l%16][k + klo + v], klo = (l>=16)?2:0 (v2f load)
// B-fragment lane l: B[n16 + l%16][k + klo + v]                    (v2f load)
// ---------------------------------------------------------------------------
template <bool GATHER, bool BIAS>
__global__ void wmma_gemm_abt(const float* __restrict__ A,
                              const float* __restrict__ Bm,
                              const float* __restrict__ bias,
                              const int* __restrict__ gidx,
                              float* __restrict__ C, int M) {
  const int lane = threadIdx.x & 31;
  const int wave = blockIdx.x * (blockDim.x >> 5) + (threadIdx.x >> 5);
  const int nBlk = D / 32;                   // 32 N-blocks of 32 cols
  const int mb = wave / nBlk;                // 64-row block index
  const int nb = wave % nBlk;                // 32-col block index
  if (mb >= M / 64) return;                  // wave-uniform
  const int l16 = lane & 15;
  const int klo = (lane >> 4) << 1;          // 0 or 2

  const float* aRow[4];
#pragma unroll
  for (int i = 0; i < 4; ++i) {
    int m = mb * 64 + i * 16 + l16;
    int src = GATHER ? ((m / NS) * L + gidx[m]) : m;
    aRow[i] = A + (size_t)src * D + klo;
  }
  const float* bRow[2];
#pragma unroll
  for (int j = 0; j < 2; ++j)
    bRow[j] = Bm + (size_t)(nb * 32 + j * 16 + l16) * D + klo;

  v8f acc[4][2] = {};
  for (int k = 0; k < D; k += 4) {
    __builtin_prefetch(bRow[0] + k + 128, 0, 0);   // global_prefetch_b8
    v2f b[2];
#pragma unroll
    for (int j = 0; j < 2; ++j) b[j] = *(const v2f*)(bRow[j] + k);
#pragma unroll
    for (int i = 0; i < 4; ++i) {
      v2f a = *(const v2f*)(aRow[i] + k);
#pragma unroll
      for (int j = 0; j < 2; ++j) acc[i][j] = wmma_f32(a, b[j], acc[i][j]);
    }
  }

  float bb[2];
#pragma unroll
  for (int j = 0; j < 2; ++j)
    bb[j] = BIAS ? bias[nb * 32 + j * 16 + l16] : 0.0f;
  const int rbase = (lane >> 4) * 8;           // C/D layout: rows r and r+8
#pragma unroll
  for (int i = 0; i < 4; ++i)
#pragma unroll
    for (int j = 0; j < 2; ++j)
#pragma unroll
      for (int r = 0; r < 8; ++r) {
        int row = mb * 64 + i * 16 + rbase + r;
        int col = nb * 32 + j * 16 + l16;
        C[(size_t)row * D + col] = acc[i][j][r] + bb[j];
      }
}

// ---------------------------------------------------------------------------
// Wd[D x D] = W[D x D] * decorr[D x D]   (B NOT transposed), 64x32 per wave.
// B-fragment lane l: B[k + klo + v][n16 + l%16]  (two strided scalar loads)
// ---------------------------------------------------------------------------
__global__ void wmma_gemm_ab(const float* __restrict__ A,
                             const float* __restrict__ Bm,
                             float* __restrict__ C) {
  const int lane = threadIdx.x & 31;
  const int wave = blockIdx.x * (blockDim.x >> 5) + (threadIdx.x >> 5);
  const int nBlk = D / 32;                   // 32
  const int mb = wave / nBlk;                // 16 blocks of 64 rows
  const int nb = wave % nBlk;
  if (mb >= D / 64) return;
  const int l16 = lane & 15;
  const int klo = (lane >> 4) << 1;

  const float* aRow[4];
#pragma unroll
  for (int i = 0; i < 4; ++i)
    aRow[i] = A + (size_t)(mb * 64 + i * 16 + l16) * D + klo;
  const float* bCol[2];
#pragma unroll
  for (int j = 0; j < 2; ++j) bCol[j] = Bm + nb * 32 + j * 16 + l16;

  v8f acc[4][2] = {};
  for (int k = 0; k < D; k += 4) {
    v2f b[2];
#pragma unroll
    for (int j = 0; j < 2; ++j) {
      b[j].x = bCol[j][(size_t)(k + klo) * D];
      b[j].y = bCol[j][(size_t)(k + klo + 1) * D];
    }
#pragma unroll
    for (int i = 0; i < 4; ++i) {
      v2f a = *(const v2f*)(aRow[i] + k);
#pragma unroll
      for (int j = 0; j < 2; ++j) acc[i][j] = wmma_f32(a, b[j], acc[i][j]);
    }
  }
  const int rbase = (lane >> 4) * 8;
#pragma unroll
  for (int i = 0; i < 4; ++i)
#pragma unroll
    for (int j = 0; j < 2; ++j)
#pragma unroll
      for (int r = 0; r < 8; ++r) {
        int row = mb * 64 + i * 16 + rbase + r;
        C[(size_t)row * D + nb * 32 + j * 16 + l16] = acc[i][j][r];
      }
}

// ---------------------------------------------------------------------------
// grad[D x D] from G = xs^T xs:  off-diag (1-k)*G/N, diag k*(G_ii/N - 1).
// grad is symmetric -> compute only upper-triangle 32x32 blocks (2x2 tiles of
// 16x16) and mirror-store.  A[M=i][K=n] = xs[n][i], B[K=n][N=j] = xs[n][j]:
// column accesses; xs (32 MB) is L2-resident (192 MB L2).
// Per k-step: 8 strided scalar loads for 4 WMMAs.
// ---------------------------------------------------------------------------
__global__ void wmma_ata_grad(const float* __restrict__ xs,
                              float* __restrict__ grad) {
  const int lane = threadIdx.x & 31;
  const int wave = blockIdx.x * (blockDim.x >> 5) + (threadIdx.x >> 5);
  const int nBlk = D / 32;                   // 32
  const int ib = wave / nBlk;
  const int jb = wave % nBlk;
  if (ib >= nBlk) return;
  if (ib > jb) return;                       // symmetric: upper triangle only
  const int l16 = lane & 15;
  const int klo = (lane >> 4) << 1;

  const float* aCol[2];
  const float* bCol[2];
#pragma unroll
  for (int t = 0; t < 2; ++t) {
    aCol[t] = xs + ib * 32 + t * 16 + l16;
    bCol[t] = xs + jb * 32 + t * 16 + l16;
  }

  v8f acc[2][2] = {};
  for (int k = 0; k < NX; k += 4) {
    v2f a[2], b[2];
#pragma unroll
    for (int t = 0; t < 2; ++t) {
      a[t].x = aCol[t][(size_t)(k + klo) * D];
      a[t].y = aCol[t][(size_t)(k + klo + 1) * D];
      b[t].x = bCol[t][(size_t)(k + klo) * D];
      b[t].y = bCol[t][(size_t)(k + klo + 1) * D];
    }
#pragma unroll
    for (int i = 0; i < 2; ++i)
#pragma unroll
      for (int j = 0; j < 2; ++j) acc[i][j] = wmma_f32(a[i], b[j], acc[i][j]);
  }

  const float invN = 1.0f / (float)NX;
  const int rbase = (lane >> 4) * 8;
#pragma unroll
  for (int i = 0; i < 2; ++i)
#pragma unroll
    for (int j = 0; j < 2; ++j)
#pragma unroll
      for (int r = 0; r < 8; ++r) {
        int row = ib * 32 + i * 16 + rbase + r;
        int col = jb * 32 + j * 16 + l16;
        float g = acc[i][j][r] * invN;
        float v = (row == col) ? KAPPA * (g - 1.0f) : (1.0f - KAPPA) * g;
        grad[(size_t)row * D + col] = v;
        grad[(size_t)col * D + row] = v;     // mirror (bitwise-identical value)
      }
}

// ---------------------------------------------------------------------------
// Per-row reduction over xs for the two losses (float4 loads):
//   acc[0] += (sum_i X2)^2   acc[1] += sum_i X2^2   acc[2] += sum_i X2
// One wave per row, lane-strided + shuffle reduction.
// ---------------------------------------------------------------------------
__global__ void loss_reduce(const float* __restrict__ xs,
                            float* __restrict__ acc3) {
  const int lane = threadIdx.x & 31;
  const int wave = blockIdx.x * (blockDim.x >> 5) + (threadIdx.x >> 5);
  if (wave >= NX) return;
  const v4f* row = (const v4f*)(xs + (size_t)wave * D);
  float s1 = 0.0f, s2 = 0.0f;
  for (int c = lane; c < D / 4; c += 32) {
    v4f v = row[c];
#pragma unroll
    for (int e = 0; e < 4; ++e) {
      float x2 = v[e] * v[e];
      s1 += x2;
      s2 += x2 * x2;
    }
  }
#pragma unroll
  for (int off = 16; off > 0; off >>= 1) {
    s1 += __shfl_xor(s1, off, 32);
    s2 += __shfl_xor(s2, off, 32);
  }
  if (lane == 0) {
    atomicAdd(&acc3[0], s1 * s1);
    atomicAdd(&acc3[1], s2);
    atomicAdd(&acc3[2], s1);
  }
}

__global__ void init_acc(float* acc3) {
  if (threadIdx.x < 3 && blockIdx.x == 0) acc3[threadIdx.x] = 0.0f;
}

__global__ void finalize_loss(const float* __restrict__ acc3,
                              float* __restrict__ out2) {
  if (threadIdx.x == 0 && blockIdx.x == 0) {
    const double denom = (double)NX * (double)D * (double)D;
    double sumS = acc3[0], trS = acc3[1], sumX2 = acc3[2];
    out2[0] = (float)((sumS - trS) / denom);                                  // correlation
    out2[1] = (float)((trS - 2.0 * sumX2 + (double)NX * (double)D) / denom);  // whitening
  }
}

// ---------------------------------------------------------------------------
extern "C" void kernel_launch(void* const* d_in, const int* in_sizes, int n_in,
                              void* d_out, int out_size, void* d_ws, size_t ws_size,
                              hipStream_t stream) {
  const float* x      = (const float*)d_in[0];  // [4, 8192, 1024]
  const float* W      = (const float*)d_in[1];  // [1024, 1024]
  const float* bias   = (const float*)d_in[2];  // [1024]
  const float* decorr = (const float*)d_in[3];  // [1024, 1024]
  const int*   sidx   = (const int*)d_in[4];    // [4, 2048]

  float* y    = (float*)d_out;                  // 33,554,432
  float* grad = y + (size_t)MY * D;             //  1,048,576
  float* loss = grad + (size_t)D * D;           //  2 scalars

  float* acc3 = (float*)d_ws;                               // 3 floats
  float* Wd   = (float*)((char*)d_ws + 256);                // 4 MB
  float* xs   = Wd + (size_t)D * D;                         // 32 MB

  const int BLK = 256;                     // 8 waves per block

  // 0) zero loss accumulators (graph-replay safe)
  hipLaunchKernelGGL(init_acc, dim3(1), dim3(32), 0, stream, acc3);

  // 1) Wd = W @ decorr           : 16*32 = 512 waves
  hipLaunchKernelGGL(wmma_gemm_ab, dim3(512 / 8), dim3(BLK), 0, stream,
                     W, decorr, Wd);

  // 2) y = x @ Wd^T + bias       : 512*32 = 16384 waves
  hipLaunchKernelGGL((wmma_gemm_abt<false, true>), dim3(16384 / 8), dim3(BLK),
                     0, stream, x, Wd, bias, (const int*)nullptr, y, MY);

  // 3) xs = gather(x) @ decorr^T : 128*32 = 4096 waves
  hipLaunchKernelGGL((wmma_gemm_abt<true, false>), dim3(4096 / 8), dim3(BLK),
                     0, stream, x, decorr, (const float*)nullptr, sidx, xs, NX);

  // 4) loss row reductions        : 8192 waves
  hipLaunchKernelGGL(loss_reduce, dim3(8192 / 8), dim3(BLK), 0, stream,
                     xs, acc3);

  // 5) grad = scaled xs^T xs      : 32*32 blocks, upper triangle active
  hipLaunchKernelGGL(wmma_ata_grad, dim3(1024 / 8), dim3(BLK), 0, stream,
                     xs, grad);

  // 6) final scalars
  hipLaunchKernelGGL(finalize_loss, dim3(1), dim3(32), 0, stream, acc3, loss);
}